// GIN_22960895164529
// MI455X (gfx1250) — compile-verified
//
#include <hip/hip_runtime.h>
#include <hip/hip_bf16.h>

#define N_NODES 100000
#define N_EDGES 1600000
#define N_GRAPHS 512
#define D_HID 128
#define D_OUT 64
#define N_TILES (N_NODES / 16)   // 6250, exact

typedef float v2f __attribute__((ext_vector_type(2)));
typedef float v8f __attribute__((ext_vector_type(8)));

// ---------------- utility kernels ----------------

__global__ void zero4_kernel(float* __restrict__ p, int n4) {
    int i = blockIdx.x * blockDim.x + threadIdx.x;
    if (i < n4) ((float4*)p)[i] = make_float4(0.f, 0.f, 0.f, 0.f);
}

__global__ void transpose_kernel(const float* __restrict__ src, float* __restrict__ dst,
                                 int rows, int cols) {
    int i = blockIdx.x * blockDim.x + threadIdx.x;
    if (i < rows * cols) {
        int r = i / cols, c = i % cols;
        dst[c * rows + r] = src[i];   // dst[n*K + k]
    }
}

// ---------------- edge aggregation: agg[dst] += h[src] ----------------
// 8 threads per edge; each thread does 4 coalesced float4 gathers + 16 f32 atomics.
__global__ void scatter_add_kernel(const float* __restrict__ h, const int* __restrict__ ei,
                                   float* __restrict__ agg) {
    int gid = blockIdx.x * blockDim.x + threadIdx.x;
    int e = gid >> 3;
    if (e >= N_EDGES) return;
    int part = gid & 7;
    int s = ei[e];
    int d = ei[N_EDGES + e];
    const float* hs = h + (size_t)s * D_HID;
    float* ad = agg + (size_t)d * D_HID;
#pragma unroll
    for (int j = 0; j < 4; ++j) {
        int c = part * 4 + j * 32;             // 8 threads cover 32 contiguous floats
        float4 v = *(const float4*)(hs + c);
        unsafeAtomicAdd(ad + c + 0, v.x);
        unsafeAtomicAdd(ad + c + 1, v.y);
        unsafeAtomicAdd(ad + c + 2, v.z);
        unsafeAtomicAdd(ad + c + 3, v.w);
    }
}

// ---------------- GIN layer GEMM: hout = act((hin+agg) @ W + b) ----------------
// 4 waves/block, each wave owns one 16-row tile of M, all 128 cols.
// A staged in LDS (stride 132 floats -> conflict-free ds_load_b64 fragments),
// B read from transposed weights (Wt[n*128 + k], contiguous v2f per fragment).
#define LROW 132
__global__ __launch_bounds__(128) void gin_gemm_relu_kernel(
    const float* hin, const float* agg, const float* __restrict__ Wt,
    const float* __restrict__ bias, float* hout, int do_relu) {
    __shared__ float lds[4 * 16 * LROW];

    int w = threadIdx.x >> 5;
    int lane = threadIdx.x & 31;
    int tile = blockIdx.x * 4 + w;
    if (tile >= N_TILES) return;               // whole-wave exit: EXEC stays all-ones
    int row0 = tile * 16;
    float* wl = lds + w * (16 * LROW);

    // ---- stage X = hin + agg (16 x 128) into LDS ----
    int r = lane >> 1, ch = lane & 1;          // 2 lanes per row, 64 cols each
    size_t gbase = (size_t)(row0 + r) * D_HID + ch * 64;
    float* lrow = wl + r * LROW + ch * 64;
#pragma unroll
    for (int j = 0; j < 16; ++j) {
        float4 a = *(const float4*)(hin + gbase + 4 * j);
        float4 g = *(const float4*)(agg + gbase + 4 * j);
        float4 s;
        s.x = a.x + g.x; s.y = a.y + g.y; s.z = a.z + g.z; s.w = a.w + g.w;
        *(float4*)(lrow + 4 * j) = s;          // wave-ordered LDS: no barrier needed
    }

    int half = lane >> 4;                       // lane group (0: lanes 0-15, 1: 16-31)
    int nl = lane & 15;
    const float* arow = wl + nl * LROW + 2 * half;  // A frag: M=nl, K = k0 + 2*half (+1)

    for (int nt = 0; nt < 8; ++nt) {
        int n = nt * 16 + nl;
        float bv = bias[n];
        v8f c = {bv, bv, bv, bv, bv, bv, bv, bv};   // bias folded into accumulator
        const float* bcol = Wt + (size_t)n * D_HID + 2 * half;
#pragma unroll
        for (int kk = 0; kk < 32; ++kk) {
            v2f a = *(const v2f*)(arow + 4 * kk);   // ds_load_b64
            v2f b = *(const v2f*)(bcol + 4 * kk);   // global_load_b64 (L2/L0-hot)
            c = __builtin_amdgcn_wmma_f32_16x16x4_f32(
                    false, a, false, b, (short)0, c, false, false);
        }
#pragma unroll
        for (int i = 0; i < 8; ++i) {
            float v = c[i];
            if (do_relu) v = fmaxf(v, 0.0f);
            hout[(size_t)(row0 + i + 8 * half) * D_HID + n] = v;
        }
    }
}

// ---------------- graph pooling ----------------
__global__ void pool_kernel(const float* __restrict__ h, const int* __restrict__ batch,
                            float* __restrict__ pooled, float* __restrict__ counts) {
    int gid = blockIdx.x * blockDim.x + threadIdx.x;
    int node = gid >> 5;
    if (node >= N_NODES) return;
    int lane = gid & 31;
    int g = batch[node];
    float4 v = *(const float4*)(h + (size_t)node * D_HID + lane * 4);
    float* pg = pooled + (size_t)g * D_HID + lane * 4;
    unsafeAtomicAdd(pg + 0, v.x);
    unsafeAtomicAdd(pg + 1, v.y);
    unsafeAtomicAdd(pg + 2, v.z);
    unsafeAtomicAdd(pg + 3, v.w);
    if (lane == 0) unsafeAtomicAdd(counts + g, 1.0f);
}

__global__ void mean_kernel(float* __restrict__ pooled, const float* __restrict__ counts) {
    int i = blockIdx.x * blockDim.x + threadIdx.x;
    if (i < N_GRAPHS * D_HID) {
        float cnt = counts[i >> 7];
        pooled[i] /= fmaxf(cnt, 1.0f);
    }
}

// ---------------- final GEMM: logits[512x64] = pooled @ wf + bf ----------------
__global__ __launch_bounds__(128) void final_gemm_kernel(
    const float* __restrict__ pooled, const float* __restrict__ WfT,
    const float* __restrict__ bf, float* __restrict__ out) {
    int w = threadIdx.x >> 5, lane = threadIdx.x & 31;
    int m0 = blockIdx.x * 16;                   // 32 blocks
    int n0 = w * 16;                            // 4 N-tiles of 16 -> D_OUT=64
    int half = lane >> 4, nl = lane & 15;
    int n = n0 + nl;
    float bv = bf[n];
    v8f c = {bv, bv, bv, bv, bv, bv, bv, bv};
    const float* ap = pooled + (size_t)(m0 + nl) * D_HID + 2 * half;
    const float* bp = WfT + (size_t)n * D_HID + 2 * half;
#pragma unroll
    for (int kk = 0; kk < 32; ++kk) {
        v2f a = *(const v2f*)(ap + 4 * kk);
        v2f b = *(const v2f*)(bp + 4 * kk);
        c = __builtin_amdgcn_wmma_f32_16x16x4_f32(
                false, a, false, b, (short)0, c, false, false);
    }
#pragma unroll
    for (int i = 0; i < 8; ++i)
        out[(size_t)(m0 + i + 8 * half) * D_OUT + n] = c[i];
}

// ---------------- launch ----------------
extern "C" void kernel_launch(void* const* d_in, const int* in_sizes, int n_in,
                              void* d_out, int out_size, void* d_ws, size_t ws_size,
                              hipStream_t stream) {
    const float* x  = (const float*)d_in[0];
    const int* ei   = (const int*)d_in[1];
    const int* batch = (const int*)d_in[2];
    const float* w1 = (const float*)d_in[3];
    const float* b1 = (const float*)d_in[4];
    const float* w2 = (const float*)d_in[5];
    const float* b2 = (const float*)d_in[6];
    const float* w3 = (const float*)d_in[7];
    const float* b3 = (const float*)d_in[8];
    const float* wf = (const float*)d_in[9];
    const float* bf = (const float*)d_in[10];
    float* out = (float*)d_out;

    char* ws = (char*)d_ws;
    const size_t HBYTES = (size_t)N_NODES * D_HID * sizeof(float);   // 51.2 MB
    float* agg  = (float*)(ws);                    // scratch aggregation buffer
    float* hbuf = (float*)(ws + HBYTES);           // node features (in-place per layer)
    float* w1t  = (float*)(ws + 2 * HBYTES);
    float* w2t  = w1t + D_HID * D_HID;
    float* w3t  = w2t + D_HID * D_HID;
    float* wft  = w3t + D_HID * D_HID;
    float* pooled = wft + D_HID * D_OUT;           // 512*128
    float* counts = pooled + N_GRAPHS * D_HID;     // 512 (contiguous with pooled)

    // transpose weights once (B fragments become contiguous 8B loads)
    transpose_kernel<<<(D_HID * D_HID + 255) / 256, 256, 0, stream>>>(w1, w1t, D_HID, D_HID);
    transpose_kernel<<<(D_HID * D_HID + 255) / 256, 256, 0, stream>>>(w2, w2t, D_HID, D_HID);
    transpose_kernel<<<(D_HID * D_HID + 255) / 256, 256, 0, stream>>>(w3, w3t, D_HID, D_HID);
    transpose_kernel<<<(D_HID * D_OUT + 255) / 256, 256, 0, stream>>>(wf, wft, D_HID, D_OUT);

    const int n4_agg = N_NODES * D_HID / 4;
    const int zgrid = (n4_agg + 255) / 256;
    const int sgrid = (N_EDGES * 8 + 255) / 256;
    const int ggrid = (N_TILES + 3) / 4;           // 1563 blocks of 4 waves

    // layer 1 (reads x, writes hbuf)
    zero4_kernel<<<zgrid, 256, 0, stream>>>(agg, n4_agg);
    scatter_add_kernel<<<sgrid, 256, 0, stream>>>(x, ei, agg);
    gin_gemm_relu_kernel<<<ggrid, 128, 0, stream>>>(x, agg, w1t, b1, hbuf, 1);

    // layer 2 (in-place on hbuf: GEMM is row-local in M)
    zero4_kernel<<<zgrid, 256, 0, stream>>>(agg, n4_agg);
    scatter_add_kernel<<<sgrid, 256, 0, stream>>>(hbuf, ei, agg);
    gin_gemm_relu_kernel<<<ggrid, 128, 0, stream>>>(hbuf, agg, w2t, b2, hbuf, 1);

    // layer 3
    zero4_kernel<<<zgrid, 256, 0, stream>>>(agg, n4_agg);
    scatter_add_kernel<<<sgrid, 256, 0, stream>>>(hbuf, ei, agg);
    gin_gemm_relu_kernel<<<ggrid, 128, 0, stream>>>(hbuf, agg, w3t, b3, hbuf, 1);

    // global mean pool
    const int n4_pool = (N_GRAPHS * D_HID + N_GRAPHS) / 4;   // pooled + counts contiguous
    zero4_kernel<<<(n4_pool + 255) / 256, 256, 0, stream>>>(pooled, n4_pool);
    pool_kernel<<<(N_NODES * 32 + 255) / 256, 256, 0, stream>>>(hbuf, batch, pooled, counts);
    mean_kernel<<<(N_GRAPHS * D_HID + 255) / 256, 256, 0, stream>>>(pooled, counts);

    // classifier
    final_gemm_kernel<<<N_GRAPHS / 16, 128, 0, stream>>>(pooled, wft, bf, out);
}